// Signature_72327249265296
// MI455X (gfx1250) — compile-verified
//
#include <hip/hip_runtime.h>

typedef __attribute__((ext_vector_type(2))) float v2f;
typedef __attribute__((ext_vector_type(8))) float v8f;

#define BATCH  64
#define LPTS   512
#define CH     10          // channels
#define NSEG   (LPTS - 1)  // 511 increments
#define CHUNKS 4
#define SEGPC  128
#define D2     100
#define D3     1000
#define D4     10000
#define SIGSZ  11110       // 10 + 100 + 1000 + 10000
#define PAD    112         // 7 * 16 (padded M/N)
#define KP     32          // padded K (actual 31)
#define RS     34          // LDS row stride (dwords): even (b64-aligned) and
                           // 34*m mod 64 distinct for m=0..15 -> bank-conflict-free
#define NT     256         // 8 waves

// ---------------------------------------------------------------------------
// Kernel 1: per-(batch, chunk) sequential Chen scan.
// Level-4 state lives in v_wmma_f32_16x16x4_f32 accumulators (49 tiles / 8
// waves); levels 1-3 live in LDS and update on VALU.
// A operand: sAop[m][k] (row stride RS). B operand stored TRANSPOSED:
// sBopT[n][k] (row stride RS) so each WMMA's (K, K+1) pair is one b64 load.
// ---------------------------------------------------------------------------
__global__ __launch_bounds__(NT) void sig_chunk_kernel(const float* __restrict__ path,
                                                       float* __restrict__ ws) {
  const int blk  = blockIdx.x;
  const int b    = blk / CHUNKS;
  const int c    = blk % CHUNKS;
  const int seg0 = c * SEGPC;
  const int nseg = min(NSEG - seg0, SEGPC);
  const int tid  = threadIdx.x;
  const int wave = tid >> 5, lane = tid & 31, half = lane >> 4, lm = lane & 15;

  __shared__ float sPath[(SEGPC + 1) * CH];
  __shared__ float sD[16];
  __shared__ float sA1[16];
  __shared__ float sA2[D2];
  __shared__ float sA3[D3];
  __shared__ float sE2[D2];
  __shared__ float sE3[D3];
  __shared__ float sAop[PAD * RS];   // A[m][k]
  __shared__ float sBopT[PAD * RS];  // B^T[n][k]

  // ---- init: stage path chunk, zero state and static-zero operand slots ----
  {
    const float* src = path + ((size_t)b * LPTS + seg0) * CH;
    for (int e = tid; e < (nseg + 1) * CH; e += NT) sPath[e] = src[e];
    if (tid < 16) { sA1[tid] = 0.f; sD[tid] = 0.f; }
    if (tid < D2) sA2[tid] = 0.f;
    for (int e = tid; e < D3; e += NT) sA3[e] = 0.f;
    for (int e = tid; e < PAD * RS; e += NT) { sAop[e] = 0.f; sBopT[e] = 0.f; }
  }
  v8f acc[7];
  #pragma unroll
  for (int i = 0; i < 7; ++i)
    #pragma unroll
    for (int v = 0; v < 8; ++v) acc[i][v] = 0.f;
  __syncthreads();
  if (tid < CH) sD[tid] = sPath[CH + tid] - sPath[tid];
  __syncthreads();

  #pragma unroll 1
  for (int s = 0; s < nseg; ++s) {
    // ---- P2: build E-levels and GEMM operands from d and old A's ----
    float rD[CH], rA1[CH];
    #pragma unroll
    for (int x = 0; x < CH; ++x) { rD[x] = sD[x]; rA1[x] = sA1[x]; }

    if (tid < D2) {
      int i = tid / CH, j = tid % CH;
      float e2 = rD[i] * rD[j] * 0.5f;
      sE2[tid] = e2;
      // B^T row n = tid: K slots 10+i (I (x) E1), 20 (E2), 21+i (I (x) d/4)
      sBopT[tid * RS + 10 + i] = rD[j];
      sBopT[tid * RS + 20]     = e2;
      sBopT[tid * RS + 21 + i] = rD[j] * 0.25f;
    }
    for (int e = tid; e < D3; e += NT) {
      int j = e / D2, k = (e / CH) % CH, l = e % CH;
      float e3 = rD[j] * rD[k] * rD[l] * (1.f / 6.f);
      sE3[e] = e3;
      sBopT[(e % D2) * RS + j] = e3;                      // K rows 0-9: E3mat (A1 (x) E3)
    }
    if (tid < D2) {
      int r = tid, i = r / CH, j = r % CH;
      sAop[r * RS + j]  = rA1[i];                         // cols 0-9: A1 (x) I
      #pragma unroll
      for (int cc = 0; cc < CH; ++cc)
        sAop[r * RS + 10 + cc] = sA3[r * CH + cc];        // cols 10-19: A3mat
      sAop[r * RS + 20] = sA2[r];                         // col 20: A2
      #pragma unroll
      for (int cc = 0; cc < CH; ++cc)
        sAop[r * RS + 21 + cc] = rD[i] * rD[j] * rD[cc] * (1.f / 6.f);  // cols 21-30: E3m2
    }
    __syncthreads();

    // ---- P3: level-4 via WMMA (f32, K=4 chained x8); level-3 on VALU ----
    #pragma unroll
    for (int ii = 0; ii < 7; ++ii) {
      int t = wave + ii * 8;            // wave-uniform => EXEC stays all-ones
      if (t < 49) {
        int ti = t / 7, tj = t % 7;
        int arow = (ti * 16 + lm) * RS;
        int brow = (tj * 16 + lm) * RS;
        #pragma unroll
        for (int kc = 0; kc < 8; ++kc) {
          int kb = kc * 4 + half * 2;   // lanes 0-15: K=kb,kb+1; lanes 16-31: +2
          v2f av, bv;
          av.x = sAop[arow + kb];
          av.y = sAop[arow + kb + 1];
          bv.x = sBopT[brow + kb];
          bv.y = sBopT[brow + kb + 1];
          acc[ii] = __builtin_amdgcn_wmma_f32_16x16x4_f32(
              false, av, false, bv, (short)0, acc[ii], false, false);
        }
      }
    }
    for (int e = tid; e < D3; e += NT)  // A3 += E3 + A1(x)E2 + A2(x)E1
      sA3[e] += sE3[e] + rA1[e / D2] * sE2[e % D2] + sA2[e / CH] * rD[e % CH];
    __syncthreads();

    // ---- P4: levels 2/1, prefetch next increment ----
    if (tid < D2) sA2[tid] += sE2[tid] + rA1[tid / CH] * rD[tid % CH];
    if (tid >= 128 && tid < 128 + CH) sA1[tid - 128] += rD[tid - 128];
    if (s + 1 < nseg && tid >= 160 && tid < 160 + CH) {
      int x = tid - 160, t2 = s + 1;
      sD[x] = sPath[(t2 + 1) * CH + x] - sPath[t2 * CH + x];
    }
    __syncthreads();
  }

  // ---- write chunk signature: [A1 | A2 | A3 | A4] ----
  {
    float* dst = ws + ((size_t)b * CHUNKS + c) * SIGSZ;
    if (tid < CH) dst[tid] = sA1[tid];
    if (tid < D2) dst[10 + tid] = sA2[tid];
    for (int e = tid; e < D3; e += NT) dst[110 + e] = sA3[e];
    #pragma unroll
    for (int ii = 0; ii < 7; ++ii) {
      int t = wave + ii * 8;
      if (t < 49) {
        int ti = t / 7, tj = t % 7;
        #pragma unroll
        for (int v = 0; v < 8; ++v) {
          int row = ti * 16 + v + half * 8;   // C/D layout: VGPR v -> M=v (lo) / v+8 (hi)
          int col = tj * 16 + lm;
          if (row < D2 && col < D2) dst[1110 + row * D2 + col] = acc[ii][v];
        }
      }
    }
  }
}

// ---------------------------------------------------------------------------
// Kernel 2: fold the CHUNKS chunk-signatures per batch via Chen's identity
// (3 combines; ~100k FLOPs per block -> pure VALU is optimal here).
// ---------------------------------------------------------------------------
__global__ __launch_bounds__(NT) void sig_combine_kernel(const float* __restrict__ ws,
                                                         float* __restrict__ out) {
  const int b = blockIdx.x;
  const int tid = threadIdx.x;
  __shared__ float s1[16];
  __shared__ float s2[D2];
  __shared__ float s3[D3];
  __shared__ float s4[D4];

  const float* base0 = ws + (size_t)b * CHUNKS * SIGSZ;
  if (tid < CH) s1[tid] = base0[tid];
  if (tid < D2) s2[tid] = base0[10 + tid];
  for (int e = tid; e < D3; e += NT) s3[e] = base0[110 + e];
  for (int e = tid; e < D4; e += NT) s4[e] = base0[1110 + e];
  __syncthreads();

  #pragma unroll 1
  for (int c = 1; c < CHUNKS; ++c) {
    const float* Bs = ws + ((size_t)b * CHUNKS + c) * SIGSZ;
    const float *B1 = Bs, *B2 = Bs + 10, *B3 = Bs + 110, *B4 = Bs + 1110;
    for (int e = tid; e < D4; e += NT)
      s4[e] += B4[e] + s1[e / D3] * B3[e % D3] + s2[e / D2] * B2[e % D2]
             + s3[e / CH] * B1[e % CH];
    __syncthreads();
    for (int e = tid; e < D3; e += NT)
      s3[e] += B3[e] + s1[e / D2] * B2[e % D2] + s2[e / CH] * B1[e % CH];
    __syncthreads();
    if (tid < D2) s2[tid] += B2[tid] + s1[tid / CH] * B1[tid % CH];
    __syncthreads();
    if (tid < CH) s1[tid] += B1[tid];
    __syncthreads();
  }

  float* dst = out + (size_t)b * SIGSZ;
  if (tid < CH) dst[tid] = s1[tid];
  if (tid < D2) dst[10 + tid] = s2[tid];
  for (int e = tid; e < D3; e += NT) dst[110 + e] = s3[e];
  for (int e = tid; e < D4; e += NT) dst[1110 + e] = s4[e];
}

extern "C" void kernel_launch(void* const* d_in, const int* in_sizes, int n_in,
                              void* d_out, int out_size, void* d_ws, size_t ws_size,
                              hipStream_t stream) {
  const float* path = (const float*)d_in[0];  // [64, 512, 10] f32
  // d_in[1] = depth (==4), hardcoded in the kernels.
  float* out = (float*)d_out;                 // [64, 11110] f32
  float* ws  = (float*)d_ws;                  // needs 64*4*11110*4 B ~= 11.4 MB

  sig_chunk_kernel<<<BATCH * CHUNKS, NT, 0, stream>>>(path, ws);
  sig_combine_kernel<<<BATCH, NT, 0, stream>>>(ws, out);
}